// MultinomialNB_2267742732999
// MI455X (gfx1250) — compile-verified
//
#include <hip/hip_runtime.h>

// ---- CDNA5 vector types -----------------------------------------------------
typedef __attribute__((ext_vector_type(2))) float        v2f;
typedef __attribute__((ext_vector_type(8))) float        v8f;
typedef __attribute__((ext_vector_type(4))) unsigned int u32x4;
typedef __attribute__((ext_vector_type(8))) int          i32x8;
typedef __attribute__((ext_vector_type(4))) int          i32x4;

// ---- Problem geometry -------------------------------------------------------
#define NROWS          16384   // B
#define NCOLS          512     // L
#define ROWS_PER_BLOCK 64      // 4 waves x 16 rows
#define WAVES_PER_BLK  4
#define THREADS_PER_BLK (WAVES_PER_BLK * 32)
#define TILE_K         64      // columns staged per TDM tile
#define NTILES         (NCOLS / TILE_K)          // 8
#define ROW_STRIDE_DW  68      // 64 data dwords + 4 pad dwords (TDM pad) -> stride % 64 == 4
#define TILE_DWORDS    (ROWS_PER_BLOCK * ROW_STRIDE_DW)

// Issue one TDM load: batch tile [ROWS_PER_BLOCK x TILE_K] (int32) -> LDS.
// D# built per CDNA5 ISA ch.8 (async_tensor): 2D tensor, data_size=4B,
// pad_interval = 64 dwords (code 5), pad_amount = 4 dwords (code 3)
// so each 64-dword row in LDS is followed by 4 pad dwords (stride 68 dwords).
__device__ __forceinline__ void issue_tile_tdm(const int* batch, int rowBlock,
                                               int tile, unsigned int lds_off) {
  unsigned long long gaddr =
      (unsigned long long)(const void*)batch +
      ((unsigned long long)rowBlock * (unsigned long long)NCOLS +
       (unsigned long long)tile * (unsigned long long)TILE_K) * 4ull;

  u32x4 g0;
  g0[0] = 1u;                                        // count=1, user descriptor
  g0[1] = lds_off;                                   // LDS byte address (tile start)
  g0[2] = (unsigned int)gaddr;                       // global_addr[31:0]
  g0[3] = (unsigned int)((gaddr >> 32) & 0x1FFFFFFull) // global_addr[56:32]
          | (2u << 30);                              // type = 2 ("image")

  i32x8 g1;
  // [15:0] wg_mask=0 | [17:16] data_size=2 (4B) | [20] pad_enable=1
  // | [24:22] pad_interval=5 (64 dw) | [31:25] pad_amount=3 (4 dw)
  g1[0] = (int)0x07520000u;
  g1[1] = (int)((unsigned)NCOLS << 16);              // tensor_dim0 = 512 (bits 79:48, low half)
  g1[2] = (int)((unsigned)NROWS << 16);              // tensor_dim1 = 16384 (bits 111:80, low half)
  g1[3] = (int)((unsigned)TILE_K << 16);             // tile_dim0 = 64 (bits 127:112)
  g1[4] = ROWS_PER_BLOCK;                            // tile_dim1 = 64, tile_dim2 = 0
  g1[5] = NCOLS;                                     // tensor_dim0_stride = 512 elements
  g1[6] = 0;                                         // stride0 hi / stride1 lo
  g1[7] = 0;

  i32x4 z4 = {0, 0, 0, 0};                           // groups 2/3 unused (2D tile)
  i32x8 z8 = {0, 0, 0, 0, 0, 0, 0, 0};               // extra group (clang-23 6-arg form)
  __builtin_amdgcn_tensor_load_to_lds(g0, g1, z4, z4, z8, /*cpol=*/0);
}

__global__ __launch_bounds__(THREADS_PER_BLK) void
nb_scores_kernel(const int* __restrict__ batch, const float* __restrict__ r,
                 const int* __restrict__ n_pos, const int* __restrict__ n_neg,
                 float* __restrict__ out) {
  __shared__ int lds_tile[2][TILE_DWORDS];

  const int lane     = threadIdx.x & 31;
  const int wave     = threadIdx.x >> 5;
  const int rowBlock = blockIdx.x * ROWS_PER_BLOCK;

  // flat pointer to LDS object: addr[31:0] is the wave-relative LDS byte offset
  const unsigned int lds_base =
      (unsigned int)(unsigned long long)(&lds_tile[0][0]);
  const unsigned int lds_buf_bytes = (unsigned int)(TILE_DWORDS * 4);

  // A-matrix (16x4 f32) lane mapping: lanes 0-15 -> M=lane, K={0,1};
  // lanes 16-31 -> M=lane-16, K={2,3}
  const int M    = lane & 15;
  const int koff = (lane >> 4) << 1;          // 0 or 2

  v8f acc  = {0.f, 0.f, 0.f, 0.f, 0.f, 0.f, 0.f, 0.f};
  v2f ones = {1.0f, 1.0f};                    // B matrix: all ones (row-sum)

  const bool leader = (wave == 0);
  if (leader) issue_tile_tdm(batch, rowBlock, 0, lds_base);

  for (int t = 0; t < NTILES; ++t) {
    if (leader) {
      if (t + 1 < NTILES) {
        issue_tile_tdm(batch, rowBlock, t + 1,
                       lds_base + (unsigned)((t + 1) & 1) * lds_buf_bytes);
        __builtin_amdgcn_s_wait_tensorcnt(1);   // tile t complete (in-order)
      } else {
        __builtin_amdgcn_s_wait_tensorcnt(0);   // last tile complete
      }
    }
    __syncthreads();                            // publish TDM-written LDS to all waves

    const int* rowLds = &lds_tile[t & 1][(wave * 16 + M) * ROW_STRIDE_DW];
#pragma unroll
    for (int kc = 0; kc < TILE_K / 4; ++kc) {
      // two tokens for this lane's (M, K-pair): 8B-aligned LDS read (ds_load_b64)
      const int tok0 = rowLds[kc * 4 + koff];
      const int tok1 = rowLds[kc * 4 + koff + 1];
      v2f a;
      a.x = r[tok0];                            // gather from L2-resident table
      a.y = r[tok1];
      // D[m][n] = sum_k A[m][k] * 1 + C[m][n]  (full f32, RNE)
      acc = __builtin_amdgcn_wmma_f32_16x16x4_f32(
          /*neg_a=*/false, a, /*neg_b=*/false, ones,
          /*c_mod=*/(short)0, acc, /*reuse_a=*/false, /*reuse_b=*/false);
    }
    __syncthreads();                            // safe to overwrite buffer t&1
  }

  // C/D layout: VGPR v -> (M=v, N=lane) for lanes 0-15, (M=v+8, N=lane-16)
  // for lanes 16-31. Column N=0 lives in lane 0 and lane 16.
  const float lp = logf((float)n_pos[0] / (float)n_neg[0]);
  const int outRow = rowBlock + wave * 16;
  if (lane == 0) {
#pragma unroll
    for (int v = 0; v < 8; ++v) out[outRow + v] = acc[v] + lp;
  } else if (lane == 16) {
#pragma unroll
    for (int v = 0; v < 8; ++v) out[outRow + 8 + v] = acc[v] + lp;
  }
}

extern "C" void kernel_launch(void* const* d_in, const int* in_sizes, int n_in,
                              void* d_out, int out_size, void* d_ws, size_t ws_size,
                              hipStream_t stream) {
  const int*   batch = (const int*)d_in[0];
  const float* r     = (const float*)d_in[1];
  const int*   n_pos = (const int*)d_in[2];
  const int*   n_neg = (const int*)d_in[3];
  float*       out   = (float*)d_out;

  dim3 grid(NROWS / ROWS_PER_BLOCK);   // 256 blocks
  dim3 block(THREADS_PER_BLK);         // 128 threads = 4 waves
  nb_scores_kernel<<<grid, block, 0, stream>>>(batch, r, n_pos, n_neg, out);
}